// ContactNet_31421980738116
// MI455X (gfx1250) — compile-verified
//
#include <hip/hip_runtime.h>
#include <stdint.h>

// ContactNet grasp construction: elementwise over 4.19M points.
// Roofline: 44 B in + 84 B out per point, ~537 MB total @ 23.3 TB/s => ~23 us
// floor; ~60 flops/point => ~0.25 GFLOP total (AI ~0.47 flop/B). Pure
// streaming, memory-bound: WMMA inapplicable. Optimize the data path only:
// wave32 256-thread blocks, b128 async global->LDS staging + b128 async
// LDS->global store for the verbatim points copy (ASYNCcnt path, no VGPR
// round trip), non-temporal b128 stores for the 64B/point grasp matrices,
// global_prefetch for the stride-12 direction streams. Bounds handling is a
// single uniform scalar branch (full tile vs ragged tail) so the hot path
// carries no exec-mask save/restore.

#define GDEPTH 0.1034f

#define AS1 __attribute__((address_space(1)))
#define AS3 __attribute__((address_space(3)))

typedef float v4f __attribute__((ext_vector_type(4)));
typedef int v4i __attribute__((vector_size(4 * sizeof(int))));

__device__ __forceinline__ AS1 v4i* to_gbl4(const void* p) {
  return (AS1 v4i*)(uintptr_t)p;
}
__device__ __forceinline__ AS3 v4i* to_lds4(const void* p) {
  return (AS3 v4i*)(uint32_t)(uintptr_t)p;
}

__global__ __launch_bounds__(256) void contactnet_grasp_kernel(
    const float* __restrict__ pts, const float* __restrict__ z1,
    const float* __restrict__ z2, const float* __restrict__ s,
    const float* __restrict__ w, float* __restrict__ out_pts,
    float* __restrict__ out_grasp, float* __restrict__ out_s,
    float* __restrict__ out_w, long long N) {
  __shared__ float lds_pts[3 * 256];  // this block's 256-point tile (3 KB)

  const int t = threadIdx.x;
  const long long i0 = (long long)blockIdx.x * 256;
  const long long i = i0 + t;
  const long long d0 = i0 * 3;  // first float of this block's points tile

  // Uniform across the block: does this block own a full 256-point tile?
  const bool full = (d0 + 768) <= 3 * N;

#if __has_builtin(__builtin_amdgcn_global_load_async_to_lds_b128) && \
    __has_builtin(__builtin_amdgcn_global_store_async_from_lds_b128)
  // --- CDNA5 async path: stage the 3KB points tile with 192 x 16B lanes ---
  if (full) {
    if (t < 192)
      __builtin_amdgcn_global_load_async_to_lds_b128(
          to_gbl4(pts + d0) + t, to_lds4(lds_pts) + t, 0, 0);
    asm volatile("s_wait_asynccnt 0" ::: "memory");
  } else {  // ragged tail: plain guarded scalar staging
    if (d0 + t < 3 * N) lds_pts[t] = pts[d0 + t];
    if (d0 + t + 256 < 3 * N) lds_pts[t + 256] = pts[d0 + t + 256];
    if (d0 + t + 512 < 3 * N) lds_pts[t + 512] = pts[d0 + t + 512];
  }
  __syncthreads();  // tile written by other waves' async copies
  // Verbatim points copy: straight LDS -> global, no VGPR round trip.
  if (full) {
    if (t < 192)
      __builtin_amdgcn_global_store_async_from_lds_b128(
          to_gbl4(out_pts + d0) + t, to_lds4(lds_pts) + t, 0, 0);
  } else {
    if (d0 + t < 3 * N) out_pts[d0 + t] = lds_pts[t];
    if (d0 + t + 256 < 3 * N) out_pts[d0 + t + 256] = lds_pts[t + 256];
    if (d0 + t + 512 < 3 * N) out_pts[d0 + t + 512] = lds_pts[t + 512];
  }
#else
  // Fallback: plain staged copy through LDS.
  if (d0 + t < 3 * N) lds_pts[t] = pts[d0 + t];
  if (d0 + t + 256 < 3 * N) lds_pts[t + 256] = pts[d0 + t + 256];
  if (d0 + t + 512 < 3 * N) lds_pts[t + 512] = pts[d0 + t + 512];
  __syncthreads();
  if (d0 + t < 3 * N) out_pts[d0 + t] = lds_pts[t];
  if (d0 + t + 256 < 3 * N) out_pts[d0 + t + 256] = lds_pts[t + 256];
  if (d0 + t + 512 < 3 * N) out_pts[d0 + t + 512] = lds_pts[t + 512];
#endif

  if (i < N) {
    // Prefetch future stride-12 stream tiles (global_prefetch_b8).
    __builtin_prefetch(z1 + 3 * i + 6144, 0, 1);
    __builtin_prefetch(z2 + 3 * i + 6144, 0, 1);

    // Contact point from LDS (stride-3 floats: conflict-free, gcd(3,64)=1).
    const float px = lds_pts[3 * t + 0];
    const float py = lds_pts[3 * t + 1];
    const float pz = lds_pts[3 * t + 2];

    // One-touch streams: non-temporal loads.
    const float ax = __builtin_nontemporal_load(z1 + 3 * i + 0);
    const float ay = __builtin_nontemporal_load(z1 + 3 * i + 1);
    const float az = __builtin_nontemporal_load(z1 + 3 * i + 2);
    const float bx = __builtin_nontemporal_load(z2 + 3 * i + 0);
    const float by = __builtin_nontemporal_load(z2 + 3 * i + 1);
    const float bz = __builtin_nontemporal_load(z2 + 3 * i + 2);
    const float sv = __builtin_nontemporal_load(s + i);
    const float wv = __builtin_nontemporal_load(w + i);

    // base_dirs = z1 / ||z1||  (== x_col)
    const float rna = __builtin_amdgcn_rsqf(ax * ax + ay * ay + az * az);
    const float xx = ax * rna, xy = ay * rna, xz = az * rna;

    // approach = (z2 - <x,z2> x) / ||z2||  ;  z_col = normalize(approach)
    const float inner = xx * bx + xy * by + xz * bz;
    const float rnb = __builtin_amdgcn_rsqf(bx * bx + by * by + bz * bz);
    float cx = (bx - inner * xx) * rnb;
    float cy = (by - inner * xy) * rnb;
    float cz = (bz - inner * xz) * rnb;
    const float rnc = __builtin_amdgcn_rsqf(cx * cx + cy * cy + cz * cz);
    const float zcx = cx * rnc, zcy = cy * rnc, zcz = cz * rnc;

    // y_col = normalize(cross(z_col, x_col))
    const float ux = zcy * xz - zcz * xy;
    const float uy = zcz * xx - zcx * xz;
    const float uz = zcx * xy - zcy * xx;
    const float rnu = __builtin_amdgcn_rsqf(ux * ux + uy * uy + uz * uz);
    const float ycx = ux * rnu, ycy = uy * rnu, ycz = uz * rnu;

    // t_col = p + (w/2) x_col - depth * z_col   (raw w, pre-ReLU)
    const float hw = 0.5f * wv;
    const float tx = px + hw * xx - GDEPTH * zcx;
    const float ty = py + hw * xy - GDEPTH * zcy;
    const float tz = pz + hw * xz - GDEPTH * zcz;

    // Row-major 4x4: row r = [x[r], y[r], z[r], t[r]], bottom = [0,0,0,1].
    v4f r0 = {xx, ycx, zcx, tx};
    v4f r1 = {xy, ycy, zcy, ty};
    v4f r2 = {xz, ycz, zcz, tz};
    v4f r3 = {0.0f, 0.0f, 0.0f, 1.0f};
    v4f* go = (v4f*)(out_grasp + i * 16);
    __builtin_nontemporal_store(r0, go + 0);
    __builtin_nontemporal_store(r1, go + 1);
    __builtin_nontemporal_store(r2, go + 2);
    __builtin_nontemporal_store(r3, go + 3);

    __builtin_nontemporal_store(1.0f / (1.0f + __expf(-sv)), out_s + i);
    __builtin_nontemporal_store(fmaxf(wv, 0.0f), out_w + i);
  }
}

extern "C" void kernel_launch(void* const* d_in, const int* in_sizes, int n_in,
                              void* d_out, int out_size, void* d_ws,
                              size_t ws_size, hipStream_t stream) {
  const float* pts = (const float*)d_in[0];
  const float* z1 = (const float*)d_in[1];
  const float* z2 = (const float*)d_in[2];
  const float* s = (const float*)d_in[3];
  const float* w = (const float*)d_in[4];
  const long long N = in_sizes[3];  // s has N elements

  float* out = (float*)d_out;
  float* out_pts = out;            // [N,3]   (== input points)
  float* out_grasp = out + 3 * N;  // [N,4,4]
  float* out_s = out + 19 * N;     // [N]
  float* out_w = out + 20 * N;     // [N]

  const int blocks = (int)((N + 255) / 256);
  contactnet_grasp_kernel<<<blocks, 256, 0, stream>>>(
      pts, z1, z2, s, w, out_pts, out_grasp, out_s, out_w, N);
}